// IGNet_62534723830004
// MI455X (gfx1250) — compile-verified
//
#include <hip/hip_runtime.h>
#include <hip/hip_bf16.h>
#include <math.h>
#include <stdint.h>

// Problem constants (from reference)
#define NVIEW 300
#define NANG  12
#define NDEP  4
#define AD    48          // NANG*NDEP
#define NB    2
#define NC    512
#define NSEED 2048
#define NGP   2000
#define KC    32          // K-chunk staged in LDS per block

typedef __attribute__((ext_vector_type(2))) float v2f;
typedef __attribute__((ext_vector_type(8))) float v8f;
typedef __attribute__((ext_vector_type(4))) unsigned int v4u;
typedef __attribute__((ext_vector_type(4))) int v4i;
typedef __attribute__((ext_vector_type(8))) int v8i;

// ----- output layout (flat f32, concatenated in reference return order) -----
#define O_VS     0u         // view_scores        (B,NS,300)   1228800
#define O_VPROT  1228800u   // vp_rot             (B,NS,3,3)     36864
#define O_GSP    1265664u   // grasp_score_pred   (B,NS,12,4)   196608
#define O_GWP    1462272u   // grasp_width_pred   (B,NS,12,4)   196608
#define O_TROT   1658880u   // top_rot            (B,NS,3,3)     36864
#define O_TSC    1695744u   // top_sc             (B,NS,12,4)   196608
#define O_TWD    1892352u   // top_wd             (B,NS,12,4)   196608
#define O_GRASP  2088960u   // graspness          (B,NS,300)   1228800
#define O_GP     3317760u   // gp                 (B,NS,3)       12288

// ===========================================================================
// GEMM: Y[b][m][n] = act( sum_k W[m][k]*X[b][k][n] + bias[m] )
// fp32 WMMA 16x16x4. block=(32,4): wave w -> m-tile blockIdx.y*4+w,
// n-range blockIdx.x*64. The KCx64 activation chunk is staged in LDS via the
// Tensor Data Mover (one 2D-tile descriptor per chunk, issued by wave 0,
// completion via s_wait_tensorcnt) and shared by all 4 waves.
// ===========================================================================
__global__ __launch_bounds__(128)
void gemm_bias_act(const float* __restrict__ W, const float* __restrict__ X,
                   const float* __restrict__ bias, float* __restrict__ Y,
                   int M, int K, int N, int relu)
{
    __shared__ float s_x[KC * 64];            // 8 KB staging tile (K-major)

    const int lane  = threadIdx.x;
    const int wv    = threadIdx.y;
    const int mtile = blockIdx.y * 4 + wv;
    const int mtiles = (M + 15) >> 4;
    const bool active = (mtile < mtiles);     // wave-uniform; barriers still hit
    const int m0 = mtile << 4;
    const int n0 = blockIdx.x << 6;
    const int b  = blockIdx.z;
    const float* __restrict__ Xb = X + (size_t)b * K * N;
    float* __restrict__ Yb = Y + (size_t)b * M * N;

    const int half = lane >> 4;               // K-half select (ISA A/B layout)
    const int l16  = lane & 15;               // M row (A) / N col (B)
    const int arow = (m0 + l16 < M) ? (m0 + l16) : (M - 1);
    const float* __restrict__ wrow = W + (size_t)arow * K;

    // ---- TDM descriptor (uniform; only global_addr changes per chunk) ----
    const uint32_t lds_off = (uint32_t)(uintptr_t)(&s_x[0]);   // LDS byte offset
    v4i zero4 = {0, 0, 0, 0};
#if defined(__clang_major__) && (__clang_major__ >= 23)
    v8i zero8 = {0, 0, 0, 0, 0, 0, 0, 0};
#endif
    v8i g1;                                    // D# group 1 (256b)
    g1[0] = (2 << 16);                         // workgroup_mask=0, data_size=4B
    g1[1] = (int)((unsigned)(N & 0xFFFF) << 16);          // tensor_dim0[15:0]
    g1[2] = (int)(((unsigned)N >> 16) & 0xFFFF) |         // tensor_dim0[31:16]
            (int)((unsigned)(K & 0xFFFF) << 16);          // tensor_dim1[15:0]
    g1[3] = (int)(((unsigned)K >> 16) & 0xFFFF) |         // tensor_dim1[31:16]
            (64 << 16);                                   // tile_dim0 = 64
    g1[4] = KC;                                           // tile_dim1 = KC
    g1[5] = N;                                            // tensor_dim0_stride
    g1[6] = 0;                                            // stride hi / dim1_stride lo
    g1[7] = 0;

    v8f acc[4] = {};
    for (int kc = 0; kc < K; kc += KC) {
        __syncthreads();                       // previous chunk fully consumed
        if (wv == 0) {                         // one TDM issue per block
            uint64_t ga = (uint64_t)(uintptr_t)(Xb + (size_t)kc * N + n0);
            v4u g0;                            // D# group 0 (128b)
            g0[0] = 1u;                        // count=1 (valid), user mode
            g0[1] = lds_off;                   // lds_addr
            g0[2] = (uint32_t)ga;              // global_addr[31:0]
            g0[3] = ((uint32_t)(ga >> 32) & 0x01FFFFFFu) | (2u << 30); // type=2
#if defined(__clang_major__) && (__clang_major__ >= 23)
            __builtin_amdgcn_tensor_load_to_lds(g0, g1, zero4, zero4, zero8, 0);
#else
            __builtin_amdgcn_tensor_load_to_lds(g0, g1, zero4, zero4, 0);
#endif
            __builtin_amdgcn_s_wait_tensorcnt(0);
        }
        __syncthreads();                       // tile visible to all waves

        if (active) {
#pragma unroll
            for (int kk = 0; kk < KC; kk += 4) {
                v2f a;                         // A 16x4: VGPR0=K(2h), VGPR1=K(2h+1)
                a.x = wrow[kc + kk + 2 * half];
                a.y = wrow[kc + kk + 2 * half + 1];
                const int r0 = (kk + 2 * half) * 64 + l16;
#pragma unroll
                for (int j = 0; j < 4; ++j) {  // B 4x16 from LDS: lane = col
                    v2f bf;
                    bf.x = s_x[r0 + 16 * j];
                    bf.y = s_x[r0 + 64 + 16 * j];
                    acc[j] = __builtin_amdgcn_wmma_f32_16x16x4_f32(
                        false, a, false, bf, (short)0, acc[j], false, false);
                }
            }
        }
    }

    if (active) {
#pragma unroll
        for (int j = 0; j < 4; ++j) {
#pragma unroll
            for (int v = 0; v < 8; ++v) {      // D: row = m0 + v + 8*half
                int row = m0 + v + 8 * half;
                if (row < M) {
                    float y = acc[j][v] + bias[row];
                    if (relu) y = fmaxf(y, 0.0f);
                    Yb[(size_t)row * N + (n0 + 16 * j + l16)] = y;
                }
            }
        }
    }
}

// ===========================================================================
// viewpoint_params_to_matrix(towards, angle=0) -> R2 (columns ax,ay,az)
// ===========================================================================
__device__ inline void vpmat(float tx, float ty, float tz, float* R) {
    float ax0 = tx, ax1 = ty, ax2 = tz;
    float ay0 = -ty, ay1 = tx, ay2 = 0.0f;
    float yn = sqrtf(ay0 * ay0 + ay1 * ay1 + ay2 * ay2);
    if (yn == 0.0f) { ay0 = 0.0f; ay1 = 1.0f; ay2 = 0.0f; }
    float an = sqrtf(ax0 * ax0 + ax1 * ax1 + ax2 * ax2);
    ax0 /= an; ax1 /= an; ax2 /= an;
    float a2 = sqrtf(ay0 * ay0 + ay1 * ay1 + ay2 * ay2);
    ay0 /= a2; ay1 /= a2; ay2 /= a2;
    float az0 = ax1 * ay2 - ax2 * ay1;
    float az1 = ax2 * ay0 - ax0 * ay2;
    float az2 = ax0 * ay1 - ax1 * ay0;
    R[0] = ax0; R[1] = ay0; R[2] = az0;
    R[3] = ax1; R[4] = ay1; R[5] = az1;
    R[6] = ax2; R[7] = ay2; R[8] = az2;
}

__global__ void init_views(float* __restrict__ views, float* __restrict__ vrot) {
    int v = blockIdx.x * blockDim.x + threadIdx.x;
    if (v >= NVIEW) return;
    const float phi = 0.61803398874989484820f;
    float Z = (2.0f * v + 1.0f) / (float)NVIEW - 1.0f;
    float r = sqrtf(fmaxf(1.0f - Z * Z, 0.0f));
    float ang = 6.28318530717958647692f * (float)v * phi;
    float X = r * cosf(ang);
    float Y = r * sinf(ang);
    views[v * 3 + 0] = X; views[v * 3 + 1] = Y; views[v * 3 + 2] = Z;
    vpmat(-X, -Y, -Z, vrot + v * 9);
}

// ===========================================================================
// Per-batch: gp_trans = R@p+t ; gv_trans = R@views ; gv_rot = R@views_rot ;
// view_inds[v] = argmin_u ||views[v]-gv_trans[u]||^2
// ===========================================================================
__global__ __launch_bounds__(256)
void batch_geom(const float* __restrict__ pose, const float* __restrict__ gpoints,
                const float* __restrict__ views, const float* __restrict__ vrot,
                float* __restrict__ gp_trans, float* __restrict__ gv_rot,
                int* __restrict__ view_inds)
{
    __shared__ float s_gv[NVIEW * 3];
    const int b = blockIdx.x, tid = threadIdx.x, nt = blockDim.x;
    const float* P = pose + b * 12;
    float R00 = P[0], R01 = P[1], R02 = P[2],  t0 = P[3];
    float R10 = P[4], R11 = P[5], R12 = P[6],  t1 = P[7];
    float R20 = P[8], R21 = P[9], R22 = P[10], t2 = P[11];

    for (int g = tid; g < NGP; g += nt) {
        float x = gpoints[(b * NGP + g) * 3 + 0];
        float y = gpoints[(b * NGP + g) * 3 + 1];
        float z = gpoints[(b * NGP + g) * 3 + 2];
        gp_trans[(b * NGP + g) * 3 + 0] = R00 * x + R01 * y + R02 * z + t0;
        gp_trans[(b * NGP + g) * 3 + 1] = R10 * x + R11 * y + R12 * z + t1;
        gp_trans[(b * NGP + g) * 3 + 2] = R20 * x + R21 * y + R22 * z + t2;
    }
    for (int v = tid; v < NVIEW; v += nt) {
        float x = views[v * 3 + 0], y = views[v * 3 + 1], z = views[v * 3 + 2];
        s_gv[v * 3 + 0] = R00 * x + R01 * y + R02 * z;
        s_gv[v * 3 + 1] = R10 * x + R11 * y + R12 * z;
        s_gv[v * 3 + 2] = R20 * x + R21 * y + R22 * z;
        const float* Vr = vrot + v * 9;
        float* O = gv_rot + (size_t)(b * NVIEW + v) * 9;
#pragma unroll
        for (int j = 0; j < 3; ++j) {
            O[0 * 3 + j] = R00 * Vr[j] + R01 * Vr[3 + j] + R02 * Vr[6 + j];
            O[1 * 3 + j] = R10 * Vr[j] + R11 * Vr[3 + j] + R12 * Vr[6 + j];
            O[2 * 3 + j] = R20 * Vr[j] + R21 * Vr[3 + j] + R22 * Vr[6 + j];
        }
    }
    __syncthreads();
    for (int v = tid; v < NVIEW; v += nt) {
        float qx = views[v * 3 + 0], qy = views[v * 3 + 1], qz = views[v * 3 + 2];
        int best = 0; float bd = 3.4e38f;
        for (int u = 0; u < NVIEW; ++u) {
            float dx = qx - s_gv[u * 3 + 0];
            float dy = qy - s_gv[u * 3 + 1];
            float dz = qz - s_gv[u * 3 + 2];
            float d = dx * dx + dy * dy + dz * dz;
            if (d < bd) { bd = d; best = u; }
        }
        view_inds[b * NVIEW + v] = best;
    }
}

// nn_inds[b,n] = argmin_g ||seed[n]-gp_trans[g]||^2 ; gp out = gp_trans[nn]
__global__ __launch_bounds__(256)
void nn_seed(const float* __restrict__ pc, const float* __restrict__ gp_trans,
             int* __restrict__ nn_inds, float* __restrict__ out_gp)
{
    __shared__ float s_gp[NGP * 3];
    const int b = blockIdx.y, tid = threadIdx.x;
    for (int i = tid; i < NGP * 3; i += 256) s_gp[i] = gp_trans[b * NGP * 3 + i];
    __syncthreads();
    const int n = blockIdx.x * 256 + tid;
    const int id = b * NSEED + n;
    float qx = pc[id * 3 + 0], qy = pc[id * 3 + 1], qz = pc[id * 3 + 2];
    int best = 0; float bd = 3.4e38f;
    for (int g = 0; g < NGP; ++g) {
        float dx = qx - s_gp[g * 3 + 0];
        float dy = qy - s_gp[g * 3 + 1];
        float dz = qz - s_gp[g * 3 + 2];
        float d = dx * dx + dy * dy + dz * dz;
        if (d < bd) { bd = d; best = g; }
    }
    nn_inds[id] = best;
    out_gp[id * 3 + 0] = s_gp[best * 3 + 0];
    out_gp[id * 3 + 1] = s_gp[best * 3 + 1];
    out_gp[id * 3 + 2] = s_gp[best * 3 + 2];
}

// transpose view scores, argmax view, vp_rot
__global__ void view_argmax(const float* __restrict__ vst, const float* __restrict__ views,
                            float* __restrict__ out_vs, float* __restrict__ out_vprot,
                            int* __restrict__ top_inds)
{
    int id = blockIdx.x * blockDim.x + threadIdx.x;
    if (id >= NB * NSEED) return;
    int b = id / NSEED, n = id % NSEED;
    const float* src = vst + (size_t)b * NVIEW * NSEED + n;
    float* dst = out_vs + (size_t)id * NVIEW;
    float best = -3.4e38f; int ti = 0;
    for (int v = 0; v < NVIEW; ++v) {
        float val = src[(size_t)v * NSEED];
        dst[v] = val;
        if (val > best) { best = val; ti = v; }   // first-max, like jnp.argmax
    }
    top_inds[id] = ti;
    float vx = views[ti * 3 + 0], vy = views[ti * 3 + 1], vz = views[ti * 3 + 2];
    float Rm[9];
    vpmat(-vx, -vy, -vz, Rm);
#pragma unroll
    for (int i = 0; i < 9; ++i) out_vprot[(size_t)id * 9 + i] = Rm[i];
}

// f2 (B,96,NS) -> score (B,NS,48) / width (B,NS,48)
__global__ void split_preds(const float* __restrict__ f2, float* __restrict__ out_sc,
                            float* __restrict__ out_wd)
{
    int id = blockIdx.x * blockDim.x + threadIdx.x;
    if (id >= NB * 96 * NSEED) return;
    int n = id % NSEED;
    int c = (id / NSEED) % 96;
    int b = id / (96 * NSEED);
    float v = f2[id];
    if (c < AD) out_sc[((size_t)(b * NSEED + n)) * AD + c] = v;
    else        out_wd[((size_t)(b * NSEED + n)) * AD + (c - AD)] = v;
}

// graspness pass 1: mean over (12,4) of (0<sc<=0.6)
__global__ void graspness1(const float* __restrict__ labels, const int* __restrict__ nn_inds,
                           const int* __restrict__ view_inds, float* __restrict__ out_g)
{
    int id = blockIdx.x * blockDim.x + threadIdx.x;
    if (id >= NB * NSEED * NVIEW) return;
    int v = id % NVIEW;
    int n = (id / NVIEW) % NSEED;
    int b = id / (NVIEW * NSEED);
    int g  = nn_inds[b * NSEED + n];
    int vi = view_inds[b * NVIEW + v];
    const float* p = labels + ((size_t)(b * NGP + g) * NVIEW + vi) * AD;
    int cnt = 0;
#pragma unroll 8
    for (int c = 0; c < AD; ++c) {
        float s = p[c];
        cnt += (s > 0.0f && s <= 0.6f) ? 1 : 0;
    }
    out_g[id] = (float)cnt * (1.0f / (float)AD);
}

// graspness pass 2: per-(b,n) row min-max normalize over 300 views
__global__ void graspness2(float* __restrict__ g)
{
    int id = blockIdx.x * blockDim.x + threadIdx.x;
    if (id >= NB * NSEED) return;
    float* row = g + (size_t)id * NVIEW;
    float mn = 3.4e38f, mx = -3.4e38f;
    for (int v = 0; v < NVIEW; ++v) { float t = row[v]; mn = fminf(mn, t); mx = fmaxf(mx, t); }
    float inv = 1.0f / (mx - mn + 1e-8f);
    for (int v = 0; v < NVIEW; ++v) row[v] = (row[v] - mn) * inv;
}

// top-view gather: top_rot, masked top_sc (pre-log), top_wd
__global__ void top_gather(const float* __restrict__ labels, const float* __restrict__ offsets,
                           const int* __restrict__ nn_inds, const int* __restrict__ view_inds,
                           const int* __restrict__ top_inds, const float* __restrict__ gv_rot,
                           float* __restrict__ out_trot, float* __restrict__ out_tsc,
                           float* __restrict__ out_twd)
{
    int id = blockIdx.x * blockDim.x + threadIdx.x;
    if (id >= NB * NSEED) return;
    int b = id / NSEED;
    int ti = top_inds[id];
    int vi = view_inds[b * NVIEW + ti];
    int g  = nn_inds[id];
    const float* Rp = gv_rot + (size_t)(b * NVIEW + vi) * 9;
#pragma unroll
    for (int i = 0; i < 9; ++i) out_trot[(size_t)id * 9 + i] = Rp[i];
    size_t base = ((size_t)(b * NGP + g) * NVIEW + vi) * AD;
#pragma unroll 8
    for (int c = 0; c < AD; ++c) {
        float sc = labels[base + c];
        float wd = offsets[base + c];
        bool m = (sc > 0.0f) && (wd <= 0.1f);
        out_twd[(size_t)id * AD + c] = wd;
        out_tsc[(size_t)id * AD + c] = m ? sc : 0.0f;
    }
}

// global u_max / u_min(>0) over top_sc
__global__ __launch_bounds__(1024)
void reduce_minmax(const float* __restrict__ tsc, float* __restrict__ scal)
{
    __shared__ float smax[1024];
    __shared__ float smin[1024];
    int tid = threadIdx.x;
    float mx = -3.4e38f, mn = 3.4e38f;
    for (int i = tid; i < NB * NSEED * AD; i += 1024) {
        float v = tsc[i];
        mx = fmaxf(mx, v);
        if (v > 0.0f) mn = fminf(mn, v);
    }
    smax[tid] = mx; smin[tid] = mn;
    __syncthreads();
    for (int s = 512; s > 0; s >>= 1) {
        if (tid < s) {
            smax[tid] = fmaxf(smax[tid], smax[tid + s]);
            smin[tid] = fminf(smin[tid], smin[tid + s]);
        }
        __syncthreads();
    }
    if (tid == 0) { scal[0] = smax[0]; scal[1] = smin[0]; }
}

__global__ void finalize_sc(float* __restrict__ tsc, const float* __restrict__ scal)
{
    int id = blockIdx.x * blockDim.x + threadIdx.x;
    if (id >= NB * NSEED * AD) return;
    float umax = scal[0], umin = scal[1];
    float denom = logf(umax / umin) + 1e-8f;
    float v = tsc[id];
    if (v > 0.0f) tsc[id] = logf(umax / v) / denom;
}

// ===========================================================================
extern "C" void kernel_launch(void* const* d_in, const int* in_sizes, int n_in,
                              void* d_out, int out_size, void* d_ws, size_t ws_size,
                              hipStream_t stream)
{
    (void)in_sizes; (void)n_in; (void)out_size; (void)ws_size;
    const float* seed_features = (const float*)d_in[0];
    const float* point_clouds  = (const float*)d_in[1];
    const float* object_pose   = (const float*)d_in[2];
    const float* grasp_points  = (const float*)d_in[3];
    const float* grasp_labels  = (const float*)d_in[4];
    const float* grasp_offsets = (const float*)d_in[5];
    const float* aw1_w = (const float*)d_in[6];  const float* aw1_b = (const float*)d_in[7];
    const float* aw2_w = (const float*)d_in[8];  const float* aw2_b = (const float*)d_in[9];
    const float* aw3_w = (const float*)d_in[10]; const float* aw3_b = (const float*)d_in[11];
    const float* sw1_w = (const float*)d_in[12]; const float* sw1_b = (const float*)d_in[13];
    const float* sw2_w = (const float*)d_in[14]; const float* sw2_b = (const float*)d_in[15];
    float* out = (float*)d_out;

    // workspace carve-up (bytes, 256-aligned); total ~23.5 MB
    char* base = (char*)d_ws;
    size_t off = 0;
    auto carve = [&](size_t bytes) { char* p = base + off; off = (off + bytes + 255) & ~(size_t)255; return p; };
    float* ws_views  = (float*)carve(NVIEW * 3 * 4);
    float* ws_vrot   = (float*)carve(NVIEW * 9 * 4);
    float* ws_a      = (float*)carve((size_t)NB * NC * NSEED * 4);   // res1, later f1
    float* ws_res    = (float*)carve((size_t)NB * NC * NSEED * 4);   // res
    float* ws_vst    = (float*)carve((size_t)NB * NVIEW * NSEED * 4);
    float* ws_f2     = (float*)carve((size_t)NB * 96 * NSEED * 4);
    float* ws_gpt    = (float*)carve((size_t)NB * NGP * 3 * 4);
    float* ws_gvrot  = (float*)carve((size_t)NB * NVIEW * 9 * 4);
    float* ws_scal   = (float*)carve(2 * 4);
    int*   ws_vinds  = (int*)carve((size_t)NB * NVIEW * 4);
    int*   ws_nn     = (int*)carve((size_t)NB * NSEED * 4);
    int*   ws_tinds  = (int*)carve((size_t)NB * NSEED * 4);

    // constant views + view rotation matrices
    init_views<<<(NVIEW + 63) / 64, 64, 0, stream>>>(ws_views, ws_vrot);

    // ---- WMMA GEMM chain (TDM-staged activations) ----
    dim3 gblk(32, 4);
    // L1: res1 = relu(aw1 @ seed + b1)
    gemm_bias_act<<<dim3(NSEED / 64, NC / 64, NB), gblk, 0, stream>>>(
        aw1_w, seed_features, aw1_b, ws_a, NC, NC, NSEED, 1);
    // L2: res = relu(aw2 @ res1 + b2)
    gemm_bias_act<<<dim3(NSEED / 64, NC / 64, NB), gblk, 0, stream>>>(
        aw2_w, ws_a, aw2_b, ws_res, NC, NC, NSEED, 1);
    // L3: view scores (B,300,NS)
    gemm_bias_act<<<dim3(NSEED / 64, (((NVIEW + 15) / 16) + 3) / 4, NB), gblk, 0, stream>>>(
        aw3_w, ws_res, aw3_b, ws_vst, NVIEW, NC, NSEED, 0);
    // L4: f1 = relu(sw1 @ res + b) (overwrites ws_a)
    gemm_bias_act<<<dim3(NSEED / 64, NC / 64, NB), gblk, 0, stream>>>(
        sw1_w, ws_res, sw1_b, ws_a, NC, NC, NSEED, 1);
    // L5: f2 = sw2 @ f1 + b  (B,96,NS)
    gemm_bias_act<<<dim3(NSEED / 64, ((96 / 16) + 3) / 4, NB), gblk, 0, stream>>>(
        sw2_w, ws_a, sw2_b, ws_f2, 96, NC, NSEED, 0);

    // ---- geometry / gathers ----
    batch_geom<<<NB, 256, 0, stream>>>(object_pose, grasp_points, ws_views, ws_vrot,
                                       ws_gpt, ws_gvrot, ws_vinds);
    nn_seed<<<dim3(NSEED / 256, NB), 256, 0, stream>>>(point_clouds, ws_gpt, ws_nn, out + O_GP);
    view_argmax<<<(NB * NSEED) / 256, 256, 0, stream>>>(ws_vst, ws_views,
                                                        out + O_VS, out + O_VPROT, ws_tinds);
    split_preds<<<(NB * 96 * NSEED) / 256, 256, 0, stream>>>(ws_f2, out + O_GSP, out + O_GWP);
    graspness1<<<(NB * NSEED * NVIEW) / 256, 256, 0, stream>>>(grasp_labels, ws_nn, ws_vinds,
                                                               out + O_GRASP);
    graspness2<<<(NB * NSEED) / 256, 256, 0, stream>>>(out + O_GRASP);
    top_gather<<<(NB * NSEED) / 256, 256, 0, stream>>>(grasp_labels, grasp_offsets, ws_nn,
                                                       ws_vinds, ws_tinds, ws_gvrot,
                                                       out + O_TROT, out + O_TSC, out + O_TWD);
    reduce_minmax<<<1, 1024, 0, stream>>>(out + O_TSC, ws_scal);
    finalize_sc<<<(NB * NSEED * AD) / 256, 256, 0, stream>>>(out + O_TSC, ws_scal);
}